// GCNConv_32701880992036
// MI455X (gfx1250) — compile-verified
//
#include <hip/hip_runtime.h>

#define N_NODES 100000
#define N_EDGES 3200000
#define D 128          // D_IN == D_OUT == 128

typedef __attribute__((ext_vector_type(2))) float v2f;
typedef __attribute__((ext_vector_type(8))) float v8f;

// ---------------------------------------------------------------------------
// Kernel 1: zero the aggregation workspace (N_NODES * D floats)
// ---------------------------------------------------------------------------
__global__ void gcn_zero_ws(float4* __restrict__ p, int n4) {
    int stride = gridDim.x * blockDim.x;
    float4 z; z.x = 0.f; z.y = 0.f; z.z = 0.f; z.w = 0.f;
    for (int i = blockIdx.x * blockDim.x + threadIdx.x; i < n4; i += stride)
        p[i] = z;
}

// ---------------------------------------------------------------------------
// Kernel 2: edge-parallel SpMM scatter.
// One wave32 per edge: lane l handles 4 consecutive features (float4 of the
// gathered X row), scaled by edge_val, accumulated with native fp32 atomics
// into agg[row].  X (51 MB) lives in the 192 MB L2, so gathers are L2 hits.
// ---------------------------------------------------------------------------
__global__ void __launch_bounds__(256)
gcn_spmm_scatter(const int* __restrict__ erow, const int* __restrict__ ecol,
                 const float* __restrict__ eval, const float* __restrict__ X,
                 float* __restrict__ agg) {
    int gid  = blockIdx.x * blockDim.x + threadIdx.x;
    int e    = gid >> 5;        // one wave per edge
    int lane = gid & 31;
    if (e >= N_EDGES) return;

    int   r = erow[e];          // uniform within wave -> broadcast load
    int   c = ecol[e];
    float v = eval[e];

    const float4* xr = (const float4*)(X + (size_t)c * D);
    float4 x = xr[lane];        // global_load_b128, coalesced across wave

    float* dst = agg + (size_t)r * D + (size_t)lane * 4;
    unsafeAtomicAdd(dst + 0, v * x.x);   // global_atomic_add_f32 (no CAS loop)
    unsafeAtomicAdd(dst + 1, v * x.y);
    unsafeAtomicAdd(dst + 2, v * x.z);
    unsafeAtomicAdd(dst + 3, v * x.w);
}

// ---------------------------------------------------------------------------
// Kernel 3: fused dense part with fp32 WMMA:
//   out = agg @ Wp^T + X @ Ws^T + b_pass + b_self
// One wave computes a 16x16 output tile with V_WMMA_F32_16X16X4_F32.
//
// fp32 A-matrix (16x4) VGPR layout: lanes 0-15 hold rows M=0..15; VGPR0 holds
// K=0 (lanes 0-15) / K=2 (lanes 16-31), VGPR1 holds K=1 / K=3.  So lane l
// loads the contiguous pair A[m][k0 + 2*(l>>4) .. +1] as a float2.
// B (4x16) mirrors this with lanes indexing columns; since B[k][j] = W[j][k]
// and W is row-major [j][k], the B fragment is also a contiguous float2 load.
// C/D layout: VGPR r -> row m0 + r + 8*(lane>>4), col j0 + (lane&15).
// ---------------------------------------------------------------------------
__global__ void __launch_bounds__(256)
gcn_gemm_wmma(const float* __restrict__ agg, const float* __restrict__ X,
              const float* __restrict__ Wp,  const float* __restrict__ bp,
              const float* __restrict__ Ws,  const float* __restrict__ bs,
              float* __restrict__ out) {
    const int lane = threadIdx.x & 31;
    const int wave = threadIdx.x >> 5;          // 0..7 -> column tile
    const int m0   = blockIdx.x * 16;           // 6250 blocks cover 100000 rows
    const int j0   = wave * 16;

    const int mrow  = m0 + (lane & 15);
    const int jcol  = j0 + (lane & 15);
    const int khalf = (lane >> 4) * 2;          // 0 or 2

    const float* arow  = agg + (size_t)mrow * D + khalf;
    const float* xrow  = X   + (size_t)mrow * D + khalf;
    const float* wprow = Wp  + (size_t)jcol * D + khalf;
    const float* wsrow = Ws  + (size_t)jcol * D + khalf;

    v8f acc = {};
    #pragma unroll 4
    for (int k = 0; k < D; k += 4) {
        v2f a  = *(const v2f*)(arow  + k);
        v2f b  = *(const v2f*)(wprow + k);
        acc = __builtin_amdgcn_wmma_f32_16x16x4_f32(
            false, a, false, b, (short)0, acc, false, false);
        v2f a2 = *(const v2f*)(xrow  + k);
        v2f b2 = *(const v2f*)(wsrow + k);
        acc = __builtin_amdgcn_wmma_f32_16x16x4_f32(
            false, a2, false, b2, (short)0, acc, false, false);
    }

    const float bias  = bp[jcol] + bs[jcol];
    const int   mbase = m0 + (lane >> 4) * 8;
    float* o = out + (size_t)mbase * D + jcol;
    #pragma unroll
    for (int vr = 0; vr < 8; ++vr)
        o[(size_t)vr * D] = acc[vr] + bias;
}

// ---------------------------------------------------------------------------
extern "C" void kernel_launch(void* const* d_in, const int* in_sizes, int n_in,
                              void* d_out, int out_size, void* d_ws, size_t ws_size,
                              hipStream_t stream) {
    const int*   erow = (const int*)  d_in[0];
    const int*   ecol = (const int*)  d_in[1];
    const float* eval = (const float*)d_in[2];
    const float* X    = (const float*)d_in[3];
    const float* Wp   = (const float*)d_in[4];
    const float* bp   = (const float*)d_in[5];
    const float* Ws   = (const float*)d_in[6];
    const float* bs   = (const float*)d_in[7];
    float*       out  = (float*)      d_out;
    float*       agg  = (float*)      d_ws;    // N_NODES * D fp32 = 51.2 MB

    // 1) agg = 0
    {
        int n4 = (N_NODES * D) / 4;
        gcn_zero_ws<<<2048, 256, 0, stream>>>((float4*)agg, n4);
    }
    // 2) agg[row] += val * X[col]   (one wave per edge, 8 edges / block)
    {
        int blocks = N_EDGES / 8;                // 3.2M edges, exact
        gcn_spmm_scatter<<<blocks, 256, 0, stream>>>(erow, ecol, eval, X, agg);
    }
    // 3) out = agg @ Wp^T + X @ Ws^T + bp + bs  (fp32 WMMA)
    {
        int blocks = N_NODES / 16;               // 6250, exact
        gcn_gemm_wmma<<<blocks, 256, 0, stream>>>(agg, X, Wp, bp, Ws, bs, out);
    }
}